// ControlNorm2D_57707180589850
// MI455X (gfx1250) — compile-verified
//
#include <hip/hip_runtime.h>

typedef __attribute__((ext_vector_type(2))) float v2f;
typedef __attribute__((ext_vector_type(4))) float v4f;
typedef __attribute__((ext_vector_type(8))) float v8f;

#define AFWD 0.999f
#define EPSV 1e-5f
#define B_  32
#define C_  256
#define HW_ 4096   // 64*64

// ---------------------------------------------------------------------------
// Kernel 1: per-(b,c) mean and variance over the 64x64 plane.
// One block per (b,c). 256 threads * 4 float4 = 4096 elements.
// Default (RT) loads intentionally warm the 192MB L2 with x for kernel 3.
// ---------------------------------------------------------------------------
__global__ __launch_bounds__(256) void cn_stats(const float* __restrict__ x,
                                                float* __restrict__ mu_g,
                                                float* __restrict__ v_g) {
    const int bc = blockIdx.x;                 // 0 .. B_*C_-1
    const v4f* xp = (const v4f*)(x + (size_t)bc * HW_);
    float s = 0.f, ss = 0.f;
#pragma unroll
    for (int i = 0; i < 4; ++i) {
        v4f t = xp[threadIdx.x + i * 256];
        s  += t.x + t.y + t.z + t.w;
        ss += t.x * t.x + t.y * t.y + t.z * t.z + t.w * t.w;
    }
    __shared__ float rs[256];
    __shared__ float rss[256];
    rs[threadIdx.x]  = s;
    rss[threadIdx.x] = ss;
    __syncthreads();
    for (int off = 128; off > 0; off >>= 1) {
        if (threadIdx.x < off) {
            rs[threadIdx.x]  += rs[threadIdx.x + off];
            rss[threadIdx.x] += rss[threadIdx.x + off];
        }
        __syncthreads();
    }
    if (threadIdx.x == 0) {
        const float inv = 1.0f / (float)HW_;
        float m = rs[0] * inv;
        mu_g[bc] = m;
        v_g[bc]  = rss[0] * inv - m * m;      // E[x^2] - mu^2
    }
}

// ---------------------------------------------------------------------------
// Kernel 2: the two lin_momentum EMA passes, done as WMMA f32 GEMMs.
//   tmp[32,16] = A[32,64] @ S[64,16],  A[i,k] = (0<=k-i<32) ? 0.999^(31-(k-i)) : 0
// One wave (32 lanes) per block, one block per group of 16 channels.
// ---------------------------------------------------------------------------
__global__ __launch_bounds__(32) void cn_ema_wmma(const float* __restrict__ mu_g,
                                                  const float* __restrict__ v_g,
                                                  const float* __restrict__ m_g,
                                                  const float* __restrict__ var_g,
                                                  const float* __restrict__ mp_g,
                                                  const float* __restrict__ vp_g,
                                                  float* __restrict__ muB_g,
                                                  float* __restrict__ invS_g) {
    __shared__ float S[64][16];        // staged seq matrix (K x N)
    __shared__ float T[32][16];        // GEMM result tmp
    __shared__ float MU[32][16], Mst[32][16], MP[32][16];
    __shared__ float V[32][16], VARs[32][16], VP[32][16];
    __shared__ float NEW[32][16];
    __shared__ float powtab[32];       // 0.999^j

    const int l    = threadIdx.x;      // 0..31, one full wave
    const int c0   = blockIdx.x * 16;  // channel group base
    const int n16  = l & 15;
    const int half = l >> 4;

    powtab[l] = powf(AFWD, (float)l);
    {
        const int base = l * C_ + c0;  // row l of each [B,C] array
        for (int n = 0; n < 16; ++n) {
            MU[l][n]   = mu_g[base + n];
            V[l][n]    = v_g[base + n];
            Mst[l][n]  = m_g[base + n];
            VARs[l][n] = var_g[base + n];
            MP[l][n]   = mp_g[base + n];
            VP[l][n]   = vp_g[base + n];
        }
    }
    __syncthreads();
    const float pow32 = powtab[31] * AFWD;        // 0.999^32
    const float omm   = 1.0f - AFWD;

    // WMMA: T = A @ S, banded A built analytically from powtab.
    auto run_gemm = [&]() {
        for (int mt = 0; mt < 2; ++mt) {          // M tiles: rows 0-15, 16-31
            v8f acc = {};
            const int mg = n16 + 16 * mt;         // global output row for A frag
#pragma unroll
            for (int kk = 0; kk < 16; ++kk) {     // K = 64 in steps of 4
                const int kb = 4 * kk + 2 * half; // this lane's K base
                v2f a, b;
                int d0 = kb - mg, d1 = kb + 1 - mg;
                int i0 = 31 - d0; i0 = i0 < 0 ? 0 : (i0 > 31 ? 31 : i0);
                int i1 = 31 - d1; i1 = i1 < 0 ? 0 : (i1 > 31 ? 31 : i1);
                float w0 = powtab[i0], w1 = powtab[i1];
                a[0] = (d0 >= 0 && d0 < 32) ? w0 : 0.0f;
                a[1] = (d1 >= 0 && d1 < 32) ? w1 : 0.0f;
                b[0] = S[kb][n16];
                b[1] = S[kb + 1][n16];
                acc = __builtin_amdgcn_wmma_f32_16x16x4_f32(
                    false, a, false, b, (short)0, acc, false, false);
            }
#pragma unroll
            for (int r = 0; r < 8; ++r)
                T[16 * mt + r + 8 * half][n16] = acc[r];
        }
    };

    // ---- pass 1: mu EMA.  seq = [m_p[1:32]; mu[0:32]], pad row 63 = 0 ----
    for (int n = 0; n < 16; ++n) {
        S[l][n]      = (l < 31) ? MP[l + 1][n] : MU[0][n];   // rows 0..31
        S[l + 32][n] = (l < 31) ? MU[l + 1][n] : 0.0f;       // rows 32..63
    }
    __syncthreads();
    run_gemm();
    __syncthreads();
    for (int n = 0; n < 16; ++n)
        NEW[l][n] = pow32 * Mst[l][n] + omm * T[l][n];
    __syncthreads();
    // stale mu, var_current, and rebuild S for pass 2 in place
    __shared__ float VC[32][16];
    for (int n = 0; n < 16; ++n) {
        float st = (l == 0) ? Mst[31][n] : NEW[l - 1][n];
        float d  = MU[l][n] - st;
        VC[l][n] = V[l][n] + AFWD * d * d;
        muB_g[l * C_ + c0 + n] = st;
    }
    __syncthreads();
    // ---- pass 2: var EMA.  seq = [var_p[1:32]; var_current[0:32]] ----
    for (int n = 0; n < 16; ++n) {
        S[l][n]      = (l < 31) ? VP[l + 1][n] : VC[0][n];
        S[l + 32][n] = (l < 31) ? VC[l + 1][n] : 0.0f;
    }
    __syncthreads();
    run_gemm();
    __syncthreads();
    for (int n = 0; n < 16; ++n)
        NEW[l][n] = pow32 * VARs[l][n] + omm * T[l][n];
    __syncthreads();
    for (int n = 0; n < 16; ++n) {
        float st = (l == 0) ? VARs[31][n] : NEW[l - 1][n];
        invS_g[l * C_ + c0 + n] = 1.0f / sqrtf(st + EPSV);
    }
}

// ---------------------------------------------------------------------------
// Kernel 3: out = (x - mu_stale) * inv_scale.  x should hit L2 (warmed by
// kernel 1); output uses non-temporal stores so it doesn't evict x.
// ---------------------------------------------------------------------------
__global__ __launch_bounds__(256) void cn_normalize(const float* __restrict__ x,
                                                    const float* __restrict__ muB,
                                                    const float* __restrict__ invS,
                                                    float* __restrict__ out) {
    const int bc = blockIdx.x;
    const float m  = muB[bc];
    const float is = invS[bc];
    const v4f* xp = (const v4f*)(x + (size_t)bc * HW_);
    v4f* op = (v4f*)(out + (size_t)bc * HW_);
#pragma unroll
    for (int i = 0; i < 4; ++i) {
        const int idx = threadIdx.x + i * 256;
        v4f t = xp[idx];
        v4f r = (t - m) * is;   // ext-vector broadcast arithmetic
        __builtin_nontemporal_store(r, op + idx);
    }
}

extern "C" void kernel_launch(void* const* d_in, const int* in_sizes, int n_in,
                              void* d_out, int out_size, void* d_ws, size_t ws_size,
                              hipStream_t stream) {
    (void)in_sizes; (void)n_in; (void)out_size; (void)ws_size;
    const float* x     = (const float*)d_in[0];
    const float* m     = (const float*)d_in[1];
    const float* var   = (const float*)d_in[2];
    const float* m_p   = (const float*)d_in[3];
    const float* var_p = (const float*)d_in[4];
    float* out = (float*)d_out;

    float* ws   = (float*)d_ws;
    float* mu   = ws;                 // [B*C]
    float* v    = ws + B_ * C_;       // [B*C]
    float* muB  = ws + 2 * B_ * C_;   // [B*C] stale mu
    float* invS = ws + 3 * B_ * C_;   // [B*C] 1/sqrt(var_stale+eps)

    cn_stats<<<B_ * C_, 256, 0, stream>>>(x, mu, v);
    cn_ema_wmma<<<C_ / 16, 32, 0, stream>>>(mu, v, m, var, m_p, var_p, muB, invS);
    cn_normalize<<<B_ * C_, 256, 0, stream>>>(x, muB, invS, out);
}